// PatchRoutingFunction_18442589569381
// MI455X (gfx1250) — compile-verified
//
#include <hip/hip_runtime.h>
#include <math.h>

typedef __attribute__((ext_vector_type(2))) float v2f;
typedef __attribute__((ext_vector_type(8))) float v8f;

#define PSZ   16
#define NE    16
#define TOPK  4
#define CCH   128
#define HH    512
#define WW    512
#define NB    8
#define FD    512
#define NPAT  1024          // patches per batch image (32*32)
#define KTOT  32768         // CCH*PSZ*PSZ

// ---------------------------------------------------------------------------
// Tiny kernel: freq_out[b][e] = dot(freq_emb[b,:], freq_w[e,:]); also aux=0.
// ---------------------------------------------------------------------------
__global__ void freq_dot_kernel(const float* __restrict__ freq_emb,
                                const float* __restrict__ freq_w,
                                float* __restrict__ freq_out,
                                float* __restrict__ aux_out) {
    int t = threadIdx.x;              // 128 threads: (b, e)
    int b = t >> 4, e = t & 15;
    const float4* fe = (const float4*)(freq_emb + (size_t)b * FD);
    const float4* fw = (const float4*)(freq_w   + (size_t)e * FD);
    float acc = 0.f;
#pragma unroll 4
    for (int i = 0; i < FD / 4; ++i) {
        float4 a = fe[i], w = fw[i];
        acc += a.x * w.x + a.y * w.y + a.z * w.z + a.w * w.w;
    }
    freq_out[t] = acc;
    if (t == 0) *aux_out = 0.f;       // eval path aux loss
}

// ---------------------------------------------------------------------------
// Main kernel: one block = 16 patches x 16 experts output tile.
// 16 waves split K (128 channels -> 8 channels/wave), each wave runs
// 512 x v_wmma_f32_16x16x4_f32, partials reduced through LDS, then the
// softmax/top-4 epilogue runs on 16 threads.
// ---------------------------------------------------------------------------
__global__ __launch_bounds__(512)
void patch_route_kernel(const float* __restrict__ x,
                        const float* __restrict__ noise,
                        const float* __restrict__ conv_w,
                        const float* __restrict__ conv_b,
                        const float* __restrict__ freq_out,
                        float* __restrict__ out) {
    __shared__ float lds_part[16 * 256];   // 16 waves x (8 vgprs x 32 lanes)
    __shared__ float lds_log[256];         // noisy logits, 16 patches x 16 experts

    const int tile  = blockIdx.x;          // 0..511
    const int b     = tile >> 6;           // batch
    const int pbase = (tile & 63) << 4;    // first patch-in-batch of this tile
    const int ph    = pbase >> 5;          // patch row (32 patches per row)
    const int pw0   = pbase & 31;          // 0 or 16

    const int tid  = threadIdx.x;
    const int wave = tid >> 5;             // 0..15 -> K split
    const int lane = tid & 31;
    const int mn   = lane & 15;            // A: patch id / B: expert id
    const int hi   = lane >> 4;            // K sub-offset (+2) for upper lanes

    // Per-lane base pointers (layout per ISA 16x16x4 f32 A/B striping).
    // A element (m=patch, k=(c,p,q)):
    //   x[((b*CCH + c)*HH + ph*PSZ + p)*WW + (pw0+m)*PSZ + q]
    const float* abase = x + ((size_t)(b * CCH) * HH + (size_t)ph * PSZ) * WW
                           + (size_t)(pw0 + mn) * PSZ + hi * 2;
    // B element (e=expert, k): conv_w[e*KTOT + k]
    const float* bbase = conv_w + (size_t)mn * KTOT + hi * 2;

    v8f acc = {0.f, 0.f, 0.f, 0.f, 0.f, 0.f, 0.f, 0.f};

    const int c0 = wave * 8;               // this wave's channel range
    for (int cc = 0; cc < 8; ++cc) {
        const float* arow = abase + (size_t)(c0 + cc) * ((size_t)HH * WW);
        const float* brow = bbase + (size_t)(c0 + cc) * (PSZ * PSZ);
#pragma unroll 4
        for (int p = 0; p < PSZ; ++p) {
            const float* ap = arow + p * WW;
            const float* bp = brow + p * PSZ;
#pragma unroll
            for (int q = 0; q < 4; ++q) {  // K = q*4 .. q*4+3 within this row
                v2f av = *(const v2f*)(ap + q * 4);
                v2f bv = *(const v2f*)(bp + q * 4);
                acc = __builtin_amdgcn_wmma_f32_16x16x4_f32(
                    /*neg_a=*/false, av, /*neg_b=*/false, bv,
                    /*c_mod=*/(short)0, acc,
                    /*reuse_a=*/false, /*reuse_b=*/false);
            }
        }
    }

    // Dump partial tile: acc[j] holds (M = j or j+8 by lane half, N = lane&15)
    {
        float* dst = lds_part + wave * 256 + lane;
#pragma unroll
        for (int j = 0; j < 8; ++j) dst[j * 32] = acc[j];
    }
    __syncthreads();

    // Cross-wave K reduction + logit assembly: thread t -> (m = t>>4, e = t&15)
    if (tid < 256) {
        const int m = tid >> 4, e = tid & 15;
        const int idx = (m & 7) * 32 + ((m & 8) << 1) + e;
        float s = 0.f;
#pragma unroll
        for (int w = 0; w < 16; ++w) s += lds_part[w * 256 + idx];
        const int pib = pbase + m;
        const float logit = s + conv_b[e] + freq_out[b * NE + e]
                          + noise[((size_t)b * NPAT + pib) * NE + e];
        lds_log[tid] = logit;              // noisy logit
    }
    __syncthreads();

    // Softmax over 16 experts + top-4 + scatter, one thread per patch.
    if (tid < 16) {
        float v[NE];
        float mx = -INFINITY;
#pragma unroll
        for (int e = 0; e < NE; ++e) { v[e] = lds_log[tid * NE + e]; mx = fmaxf(mx, v[e]); }
        float sum = 0.f;
#pragma unroll
        for (int e = 0; e < NE; ++e) { v[e] = __expf(v[e] - mx); sum += v[e]; }
        const float inv = 1.f / sum;
#pragma unroll
        for (int e = 0; e < NE; ++e) v[e] *= inv;

        int   idxs[TOPK];
        float vals[TOPK];
        unsigned taken = 0u;
        for (int k = 0; k < TOPK; ++k) {   // strict '>' => lowest index wins ties
            float best = -1.f; int bi = 0;
            for (int e = 0; e < NE; ++e)
                if (!((taken >> e) & 1u) && v[e] > best) { best = v[e]; bi = e; }
            taken |= 1u << bi;
            idxs[k] = bi; vals[k] = best;
        }

        const size_t g = (size_t)b * NPAT + pbase + tid;   // global patch index
        float* gates = out + g * NE;
#pragma unroll
        for (int e = 0; e < NE; ++e) gates[e] = ((taken >> e) & 1u) ? v[e] : 0.f;

        float* oi = out + (size_t)NB * NPAT * NE + g * TOPK;
        float* ov = out + (size_t)NB * NPAT * NE + (size_t)NB * NPAT * TOPK + g * TOPK;
#pragma unroll
        for (int k = 0; k < TOPK; ++k) { oi[k] = (float)idxs[k]; ov[k] = vals[k]; }
    }
}

// ---------------------------------------------------------------------------
extern "C" void kernel_launch(void* const* d_in, const int* in_sizes, int n_in,
                              void* d_out, int out_size, void* d_ws, size_t ws_size,
                              hipStream_t stream) {
    const float* x        = (const float*)d_in[0];
    const float* freq_emb = (const float*)d_in[1];
    const float* noise    = (const float*)d_in[2];
    const float* conv_w   = (const float*)d_in[3];
    const float* conv_b   = (const float*)d_in[4];
    const float* freq_w   = (const float*)d_in[5];
    float* out = (float*)d_out;
    float* ws  = (float*)d_ws;                         // 128 floats: freq table
    float* aux = out + (size_t)NB * NPAT * NE
                     + 2 * (size_t)NB * NPAT * TOPK;   // element 196608

    freq_dot_kernel<<<1, 128, 0, stream>>>(freq_emb, freq_w, ws, aux);
    patch_route_kernel<<<512, 512, 0, stream>>>(x, noise, conv_w, conv_b, ws, out);
}